// CKA_loss_37211596653351
// MI455X (gfx1250) — compile-verified
//
#include <hip/hip_runtime.h>
#include <hip/hip_bf16.h>
#include <math.h>

// CKA loss, factorized:  cX = (HX)(HX)^T  =>
//   hsic = ||X^T Y - sx sy^T/n||_F^2 ,  varx = ||X^T X - sx sx^T/n||_F , ...
// Split-K fp32 GEMMs (64x1024 . 1024x64) via V_WMMA_F32_16X16X4_F32, with
// double-buffered GLOBAL_LOAD_ASYNC_TO_LDS_B128 staging (ASYNCcnt pipelined).

typedef float v2f __attribute__((ext_vector_type(2)));
typedef float v8f __attribute__((ext_vector_type(8)));

#define N_ROWS 1024
#define DDIM   64
#define KC     64          // K-chunk (sync fallback path)
#define KCA    32          // K-chunk per buffer (async double-buffered path)
#define LDSW   72          // LDS row stride (floats): bank=(8k+m+16t)%64 -> conflict-free
#define NPAIR  40          // L*B = 5*8
#define MATSZ  4096        // 64*64
#define SPLIT_STRIDE (3 * MATSZ + 2 * DDIM)   // floats per (pair,split) record

#if defined(__AMDGCN__) && __has_builtin(__builtin_amdgcn_global_load_async_to_lds_b128)
#define HAVE_ASYNC_LDS 1
typedef int v4i __attribute__((ext_vector_type(4)));
typedef __attribute__((address_space(1))) v4i *g4ptr_t;   // global int4*
typedef __attribute__((address_space(3))) v4i *l4ptr_t;   // LDS int4*
#if __has_builtin(__builtin_amdgcn_s_wait_asynccnt)
#define WAIT_ASYNC(n) __builtin_amdgcn_s_wait_asynccnt((unsigned short)(n))
#else
#define WAIT_ASYNC(n) asm volatile("s_wait_asynccnt %0" :: "i"(n) : "memory")
#endif
#else
#define HAVE_ASYNC_LDS 0
#endif

// ---------------------------------------------------------------------------
// Kernel A: split-K partial GEMMs + column sums (one global pass)
// ---------------------------------------------------------------------------
__global__ __launch_bounds__(512)
void cka_split_kernel(const float* __restrict__ teacher,
                      const float* __restrict__ student,
                      float* __restrict__ partials,
                      int nsplit)
{
    const int pair  = blockIdx.x / nsplit;
    const int split = blockIdx.x - pair * nsplit;
    const int tid   = threadIdx.x;
    const int lane  = tid & 31;
    const int wave  = tid >> 5;                 // 0..15

    const float* __restrict__ X = teacher + (size_t)pair * (N_ROWS * DDIM);
    const float* __restrict__ Y = student + (size_t)pair * (N_ROWS * DDIM);

    const int ti    = wave >> 2;
    const int tj    = wave & 3;
    const int mcol  = ti * 16 + (lane & 15);    // A-fragment column
    const int ncol  = tj * 16 + (lane & 15);    // B-fragment column
    const int khalf = (lane >> 4) << 1;         // lanes 0-15: K+0/1, 16-31: K+2/3

    __shared__ float sx[DDIM];
    __shared__ float sy[DDIM];

    v8f accXX = {}; v8f accXY = {}; v8f accYY = {};

#if HAVE_ASYNC_LDS
    // ---------- double-buffered async staging ----------
    __shared__ float Xs[2][KCA * LDSW];
    __shared__ float Ys[2][KCA * LDSW];

    const int rows   = N_ROWS / nsplit;
    const int row0   = split * rows;
    const int chunks = rows / KCA;

    const int rr  = tid >> 4;                   // 0..31 row in chunk
    const int cc4 = (tid & 15) << 2;            // column group

    auto issue = [&](int chunk, int buf) {
        const float* gx = X + (size_t)(row0 + chunk * KCA + rr) * DDIM + cc4;
        const float* gy = Y + (size_t)(row0 + chunk * KCA + rr) * DDIM + cc4;
        __builtin_amdgcn_global_load_async_to_lds_b128(
            (g4ptr_t)gx, (l4ptr_t)&Xs[buf][rr * LDSW + cc4], 0, 0);
        __builtin_amdgcn_global_load_async_to_lds_b128(
            (g4ptr_t)gy, (l4ptr_t)&Ys[buf][rr * LDSW + cc4], 0, 0);
    };

    float psx = 0.0f, psy = 0.0f;               // column-sum partial (col = tid&63)
    const int csc = tid & 63;
    const int csg = tid >> 6;                   // 0..7

    issue(0, 0);
    for (int c = 0; c < chunks; ++c) {
        const int buf = c & 1;
        if (c + 1 < chunks) {
            issue(c + 1, buf ^ 1);
            WAIT_ASYNC(2);                      // chunk c landed; c+1 may fly
        } else {
            WAIT_ASYNC(0);
        }
        __syncthreads();

        #pragma unroll
        for (int j = 0; j < KCA / 8; ++j) {     // column sums from LDS
            const int r = csg + (j << 3);
            psx += Xs[buf][r * LDSW + csc];
            psy += Ys[buf][r * LDSW + csc];
        }

        #pragma unroll 4
        for (int kk = 0; kk < KCA; kk += 4) {
            const int kA = kk + khalf;
            v2f ax, bx, ay, by;
            ax.x = Xs[buf][kA * LDSW + mcol];   ax.y = Xs[buf][(kA + 1) * LDSW + mcol];
            bx.x = Xs[buf][kA * LDSW + ncol];   bx.y = Xs[buf][(kA + 1) * LDSW + ncol];
            ay.x = Ys[buf][kA * LDSW + mcol];   ay.y = Ys[buf][(kA + 1) * LDSW + mcol];
            by.x = Ys[buf][kA * LDSW + ncol];   by.y = Ys[buf][(kA + 1) * LDSW + ncol];
            accXX = __builtin_amdgcn_wmma_f32_16x16x4_f32(false, ax, false, bx,
                                                          (short)0, accXX, false, false);
            accXY = __builtin_amdgcn_wmma_f32_16x16x4_f32(false, ax, false, by,
                                                          (short)0, accXY, false, false);
            accYY = __builtin_amdgcn_wmma_f32_16x16x4_f32(false, ay, false, by,
                                                          (short)0, accYY, false, false);
        }
        __syncthreads();                        // release buf before reuse
    }

    if (tid < DDIM) { sx[tid] = 0.0f; sy[tid] = 0.0f; }
    __syncthreads();
    atomicAdd(&sx[csc], psx);
    atomicAdd(&sy[csc], psy);
    __syncthreads();
#else
    // ---------- synchronous fallback staging (proven path) ----------
    __shared__ float Xs[KC * LDSW];
    __shared__ float Ys[KC * LDSW];

    const int rows   = N_ROWS / nsplit;
    const int row0   = split * rows;
    const int chunks = rows / KC;

    float psx0 = 0.f, psx1 = 0.f, psx2 = 0.f, psx3 = 0.f;
    float psy0 = 0.f, psy1 = 0.f, psy2 = 0.f, psy3 = 0.f;

    for (int chunk = 0; chunk < chunks; ++chunk) {
        #pragma unroll
        for (int i = 0; i < 2; ++i) {
            const int f  = tid + i * 512;
            const int r  = f >> 4;
            const int c4 = (f & 15) << 2;
            const float4 vx = *(const float4*)(X + (size_t)(row0 + chunk * KC + r) * DDIM + c4);
            const float4 vy = *(const float4*)(Y + (size_t)(row0 + chunk * KC + r) * DDIM + c4);
            psx0 += vx.x; psx1 += vx.y; psx2 += vx.z; psx3 += vx.w;
            psy0 += vy.x; psy1 += vy.y; psy2 += vy.z; psy3 += vy.w;
            *(float4*)(&Xs[r * LDSW + c4]) = vx;
            *(float4*)(&Ys[r * LDSW + c4]) = vy;
        }
        __syncthreads();
        #pragma unroll 4
        for (int kk = 0; kk < KC; kk += 4) {
            const int kA = kk + khalf;
            v2f ax, bx, ay, by;
            ax.x = Xs[kA * LDSW + mcol];       ax.y = Xs[(kA + 1) * LDSW + mcol];
            bx.x = Xs[kA * LDSW + ncol];       bx.y = Xs[(kA + 1) * LDSW + ncol];
            ay.x = Ys[kA * LDSW + mcol];       ay.y = Ys[(kA + 1) * LDSW + mcol];
            by.x = Ys[kA * LDSW + ncol];       by.y = Ys[(kA + 1) * LDSW + ncol];
            accXX = __builtin_amdgcn_wmma_f32_16x16x4_f32(false, ax, false, bx,
                                                          (short)0, accXX, false, false);
            accXY = __builtin_amdgcn_wmma_f32_16x16x4_f32(false, ax, false, by,
                                                          (short)0, accXY, false, false);
            accYY = __builtin_amdgcn_wmma_f32_16x16x4_f32(false, ay, false, by,
                                                          (short)0, accYY, false, false);
        }
        __syncthreads();
    }

    if (tid < DDIM) { sx[tid] = 0.0f; sy[tid] = 0.0f; }
    __syncthreads();
    {
        const int c4 = (tid & 15) << 2;
        atomicAdd(&sx[c4 + 0], psx0); atomicAdd(&sx[c4 + 1], psx1);
        atomicAdd(&sx[c4 + 2], psx2); atomicAdd(&sx[c4 + 3], psx3);
        atomicAdd(&sy[c4 + 0], psy0); atomicAdd(&sy[c4 + 1], psy1);
        atomicAdd(&sy[c4 + 2], psy2); atomicAdd(&sy[c4 + 3], psy3);
    }
    __syncthreads();
#endif

    // ---------- emit (pair,split) record ----------
    float* __restrict__ out = partials + (size_t)blockIdx.x * SPLIT_STRIDE;
    const int mhalf = (lane >> 4) << 3;          // C layout: lanes 16-31 hold M+8
    #pragma unroll
    for (int v = 0; v < 8; ++v) {
        const int gm = ti * 16 + v + mhalf;
        const int gn = tj * 16 + (lane & 15);
        out[0 * MATSZ + gm * DDIM + gn] = accXX[v];
        out[1 * MATSZ + gm * DDIM + gn] = accXY[v];
        out[2 * MATSZ + gm * DDIM + gn] = accYY[v];
    }
    if (tid < DDIM) {
        out[3 * MATSZ + tid]        = sx[tid];
        out[3 * MATSZ + DDIM + tid] = sy[tid];
    }
}

// ---------------------------------------------------------------------------
// Kernel B: sum splits, rank-1 correction, Frobenius norms -> ratio per pair
// ---------------------------------------------------------------------------
__global__ __launch_bounds__(64)
void cka_reduce_kernel(const float* __restrict__ partials,
                       float* __restrict__ ratios,
                       int nsplit)
{
    __shared__ float tsx[DDIM], tsy[DDIM];
    __shared__ float rh[DDIM], ra[DDIM], rb[DDIM];

    const int pair = blockIdx.x;
    const int t    = threadIdx.x;                // 0..63, owns column n = t
    const float* __restrict__ base = partials + (size_t)pair * nsplit * SPLIT_STRIDE;

    float axs = 0.0f, ays = 0.0f;
    for (int s = 0; s < nsplit; ++s) {
        axs += base[(size_t)s * SPLIT_STRIDE + 3 * MATSZ + t];
        ays += base[(size_t)s * SPLIT_STRIDE + 3 * MATSZ + DDIM + t];
    }
    tsx[t] = axs; tsy[t] = ays;
    __syncthreads();

    const float invn = 1.0f / (float)N_ROWS;
    const float sxn = tsx[t], syn = tsy[t];
    float h = 0.0f, a2 = 0.0f, b2 = 0.0f;
    for (int m = 0; m < DDIM; ++m) {             // coalesced: thread = column
        float sxx = 0.0f, sxy = 0.0f, syy = 0.0f;
        for (int s = 0; s < nsplit; ++s) {
            const float* p = base + (size_t)s * SPLIT_STRIDE + m * DDIM + t;
            sxx += p[0 * MATSZ];
            sxy += p[1 * MATSZ];
            syy += p[2 * MATSZ];
        }
        const float sxm = tsx[m], sym = tsy[m];
        const float gxy = sxy - sxm * syn * invn;
        const float gxx = sxx - sxm * sxn * invn;
        const float gyy = syy - sym * syn * invn;
        h  += gxy * gxy;
        a2 += gxx * gxx;
        b2 += gyy * gyy;
    }
    rh[t] = h; ra[t] = a2; rb[t] = b2;
    __syncthreads();
    if (t == 0) {
        float H = 0.0f, A = 0.0f, Bv = 0.0f;
        for (int i = 0; i < DDIM; ++i) { H += rh[i]; A += ra[i]; Bv += rb[i]; }
        ratios[pair] = fabsf(H / (sqrtf(A) * sqrtf(Bv)));
    }
}

// ---------------------------------------------------------------------------
// Kernel C: final scalar loss
// ---------------------------------------------------------------------------
__global__ void cka_final_kernel(const float* __restrict__ ratios,
                                 float* __restrict__ out)
{
    if (threadIdx.x == 0 && blockIdx.x == 0) {
        float acc = 0.0f;
        #pragma unroll
        for (int l = 0; l < 5; ++l) {
            float s = 0.0f;
            #pragma unroll
            for (int b = 0; b < 8; ++b) s += ratios[l * 8 + b];
            acc += -logf(s * 0.125f + 1e-8f);
        }
        out[0] = acc * 0.2f;
    }
}

// ---------------------------------------------------------------------------
// Fallback (fused per-pair kernel) for tiny workspaces: needs only 40 floats
// ---------------------------------------------------------------------------
__global__ __launch_bounds__(512)
void cka_fused_kernel(const float* __restrict__ teacher,
                      const float* __restrict__ student,
                      float* __restrict__ ratios)
{
    __shared__ float Xs[KC * LDSW];
    __shared__ float Ys[KC * LDSW];
    __shared__ float sx[DDIM];
    __shared__ float sy[DDIM];
    __shared__ float red[48];

    const int blk  = blockIdx.x;
    const int tid  = threadIdx.x;
    const int lane = tid & 31;
    const int wave = tid >> 5;

    const float* __restrict__ X = teacher + (size_t)blk * (N_ROWS * DDIM);
    const float* __restrict__ Y = student + (size_t)blk * (N_ROWS * DDIM);

    const int ti    = wave >> 2;
    const int tj    = wave & 3;
    const int mcol  = ti * 16 + (lane & 15);
    const int ncol  = tj * 16 + (lane & 15);
    const int khalf = (lane >> 4) << 1;

    v8f accXX = {}; v8f accXY = {}; v8f accYY = {};
    float psx0 = 0.f, psx1 = 0.f, psx2 = 0.f, psx3 = 0.f;
    float psy0 = 0.f, psy1 = 0.f, psy2 = 0.f, psy3 = 0.f;

    for (int chunk = 0; chunk < N_ROWS / KC; ++chunk) {
        #pragma unroll
        for (int i = 0; i < 2; ++i) {
            const int f  = tid + i * 512;
            const int r  = f >> 4;
            const int c4 = (f & 15) << 2;
            const float4 vx = *(const float4*)(X + (size_t)(chunk * KC + r) * DDIM + c4);
            const float4 vy = *(const float4*)(Y + (size_t)(chunk * KC + r) * DDIM + c4);
            psx0 += vx.x; psx1 += vx.y; psx2 += vx.z; psx3 += vx.w;
            psy0 += vy.x; psy1 += vy.y; psy2 += vy.z; psy3 += vy.w;
            *(float4*)(&Xs[r * LDSW + c4]) = vx;
            *(float4*)(&Ys[r * LDSW + c4]) = vy;
        }
        __syncthreads();
        #pragma unroll 4
        for (int kk = 0; kk < KC; kk += 4) {
            const int kA = kk + khalf;
            v2f ax, bx, ay, by;
            ax.x = Xs[kA * LDSW + mcol];       ax.y = Xs[(kA + 1) * LDSW + mcol];
            bx.x = Xs[kA * LDSW + ncol];       bx.y = Xs[(kA + 1) * LDSW + ncol];
            ay.x = Ys[kA * LDSW + mcol];       ay.y = Ys[(kA + 1) * LDSW + mcol];
            by.x = Ys[kA * LDSW + ncol];       by.y = Ys[(kA + 1) * LDSW + ncol];
            accXX = __builtin_amdgcn_wmma_f32_16x16x4_f32(false, ax, false, bx,
                                                          (short)0, accXX, false, false);
            accXY = __builtin_amdgcn_wmma_f32_16x16x4_f32(false, ax, false, by,
                                                          (short)0, accXY, false, false);
            accYY = __builtin_amdgcn_wmma_f32_16x16x4_f32(false, ay, false, by,
                                                          (short)0, accYY, false, false);
        }
        __syncthreads();
    }

    if (tid < DDIM) { sx[tid] = 0.0f; sy[tid] = 0.0f; }
    __syncthreads();
    {
        const int c4 = (tid & 15) << 2;
        atomicAdd(&sx[c4 + 0], psx0); atomicAdd(&sx[c4 + 1], psx1);
        atomicAdd(&sx[c4 + 2], psx2); atomicAdd(&sx[c4 + 3], psx3);
        atomicAdd(&sy[c4 + 0], psy0); atomicAdd(&sy[c4 + 1], psy1);
        atomicAdd(&sy[c4 + 2], psy2); atomicAdd(&sy[c4 + 3], psy3);
    }
    __syncthreads();

    const float invn  = 1.0f / (float)N_ROWS;
    const int   mhalf = (lane >> 4) << 3;
    float h = 0.0f, a2 = 0.0f, b2 = 0.0f;
    #pragma unroll
    for (int v = 0; v < 8; ++v) {
        const int gm = ti * 16 + v + mhalf;
        const int gn = tj * 16 + (lane & 15);
        const float sxm = sx[gm], sxn = sx[gn];
        const float sym = sy[gm], syn = sy[gn];
        const float gxy = accXY[v] - sxm * syn * invn;
        const float gxx = accXX[v] - sxm * sxn * invn;
        const float gyy = accYY[v] - sym * syn * invn;
        h  += gxy * gxy; a2 += gxx * gxx; b2 += gyy * gyy;
    }
    #pragma unroll
    for (int off = 16; off > 0; off >>= 1) {
        h  += __shfl_xor(h,  off, 32);
        a2 += __shfl_xor(a2, off, 32);
        b2 += __shfl_xor(b2, off, 32);
    }
    if (lane == 0) { red[wave] = h; red[16 + wave] = a2; red[32 + wave] = b2; }
    __syncthreads();
    if (tid == 0) {
        float H = 0.0f, A = 0.0f, Bv = 0.0f;
        for (int w = 0; w < 16; ++w) { H += red[w]; A += red[16 + w]; Bv += red[32 + w]; }
        ratios[blk] = fabsf(H / (sqrtf(A) * sqrtf(Bv)));
    }
}

// ---------------------------------------------------------------------------
extern "C" void kernel_launch(void* const* d_in, const int* in_sizes, int n_in,
                              void* d_out, int out_size, void* d_ws, size_t ws_size,
                              hipStream_t stream)
{
    const float* teacher = (const float*)d_in[0];   // [5,8,1024,64] fp32
    const float* student = (const float*)d_in[1];   // [5,8,1024,64] fp32
    float* ws = (float*)d_ws;

    // choose split factor deterministically from ws_size
    const size_t recf = (size_t)SPLIT_STRIDE;       // floats per record
    int nsplit = 8;
    while (nsplit > 1 &&
           ((size_t)NPAIR * nsplit * recf + NPAIR) * sizeof(float) > ws_size)
        nsplit >>= 1;

    if (((size_t)NPAIR * nsplit * recf + NPAIR) * sizeof(float) <= ws_size) {
        float* partials = ws;
        float* ratios   = ws + (size_t)NPAIR * nsplit * recf;
        cka_split_kernel<<<NPAIR * nsplit, 512, 0, stream>>>(teacher, student, partials, nsplit);
        cka_reduce_kernel<<<NPAIR, 64, 0, stream>>>(partials, ratios, nsplit);
        cka_final_kernel<<<1, 32, 0, stream>>>(ratios, (float*)d_out);
    } else {
        // tiny workspace fallback: fused per-pair kernel (40 floats of ws)
        float* ratios = ws;
        cka_fused_kernel<<<NPAIR, 512, 0, stream>>>(teacher, student, ratios);
        cka_final_kernel<<<1, 32, 0, stream>>>(ratios, (float*)d_out);
    }
}